// MyRGCNConvHistory_83932250898805
// MI455X (gfx1250) — compile-verified
//
#include <hip/hip_runtime.h>
#include <hip/hip_bf16.h>

typedef __attribute__((ext_vector_type(2))) float v2f;
typedef __attribute__((ext_vector_type(8))) float v8f;

#define IN_C   64
#define HID_C  64
#define N_REL  8
#define TILE_M 16
#define SROW   68   // padded LDS row stride (floats): stride-4-banks per lane -> conflict free

// Wt[r][n][k] = W[r][k][n]  (so B-operand K-pairs are contiguous 8B loads)
__global__ void wt_transpose_kernel(const float* __restrict__ W,
                                    float* __restrict__ Wt, int total) {
  int i = blockIdx.x * blockDim.x + threadIdx.x;
  if (i < total) {
    int r = i >> 12;          // / (64*64)
    int k = (i >> 6) & 63;
    int n = i & 63;
    Wt[(r << 12) + (n << 6) + k] = W[i];
  }
}

__global__ __launch_bounds__(128) void rgcn_fused_kernel(
    const float* __restrict__ x,
    const float* __restrict__ Wt,     // [R][out][in]
    const int*   __restrict__ ptr,
    const int*   __restrict__ idx,
    const int*   __restrict__ et,
    const int*   __restrict__ hmap,
    const float* __restrict__ hbuf,
    const int*   __restrict__ hsize_p,
    float* __restrict__ out,
    int n_node)
{
  __shared__ float S[N_REL * TILE_M * SROW];   // ~34 KB: per-relation neighbor sums
  __shared__ float s_deg[TILE_M];
  __shared__ int   s_map[TILE_M];

  const int tid   = threadIdx.x;
  const int lane  = tid & 31;
  const int wave  = tid >> 5;          // 0..3
  const int node0 = blockIdx.x * TILE_M;

  if (tid < TILE_M) {
    int n  = node0 + tid;
    int nc = (n < n_node) ? n : (n_node - 1);
    s_deg[tid] = (float)(ptr[nc + 1] - ptr[nc]);
    s_map[tid] = hmap[nc];
  }

  // ---- phase 1: gather x rows, bucket-sum by relation into LDS ----
  // All per-edge control values are wave-uniform: force scalar (SGPR) form so
  // idx/et become scalar loads, the gather uses saddr addressing, and the
  // relation dispatch becomes scalar compare + uniform branch (1 pk-add/edge).
  for (int nn = 0; nn < 4; ++nn) {
    const int local = wave * 4 + nn;
    int n = node0 + local;
    if (n >= n_node) n = n_node - 1;          // tail: compute garbage, stores masked later
    const int e0 = __builtin_amdgcn_readfirstlane(ptr[n]);
    const int e1 = __builtin_amdgcn_readfirstlane(ptr[n + 1]);
    v2f acc[N_REL];
#pragma unroll
    for (int r = 0; r < N_REL; ++r) acc[r] = (v2f)0.0f;
    for (int e = e0; e < e1; ++e) {
      const int src = __builtin_amdgcn_readfirstlane(idx[e]);
      const int t   = __builtin_amdgcn_readfirstlane(et[e]);
      const v2f v = *(const v2f*)(x + (size_t)src * IN_C + 2 * lane);  // coalesced 256B/wave
#pragma unroll
      for (int r = 0; r < N_REL; ++r) {
        if (t == r) acc[r] += v;              // scalar-branch guarded v_pk_add_f32
      }
    }
#pragma unroll
    for (int r = 0; r < N_REL; ++r) {
      *(v2f*)(S + (r * TILE_M + local) * SROW + 2 * lane) = acc[r];
    }
  }
  __syncthreads();

  // ---- phase 2: out_tile[16x64] = sum_r S_r[16x64] @ W_r[64x64], f32 WMMA ----
  // A 16x4 layout: lanes 0-15 row M=lane, K={k0,k0+1}; lanes 16-31 row M=lane-16, K={k0+2,k0+3}
  // B 4x16 layout: lanes 0-15 N=lane, K={k0,k0+1}; lanes 16-31 N=lane-16, K={k0+2,k0+3}
  const int ncol = lane & 15;
  const int hi   = lane >> 4;
  const int n0   = wave * 16;          // this wave's output column tile
  v8f acc = {};

#pragma unroll 2
  for (int r = 0; r < N_REL; ++r) {
    const float* Sr = S + r * TILE_M * SROW;
    const float* Br = Wt + (r << 12) + ((n0 + ncol) << 6);
#pragma unroll
    for (int j = 0; j < 16; ++j) {
      const int k = 4 * j + 2 * hi;
      const v2f A = *(const v2f*)(Sr + ncol * SROW + k);
      const v2f B = *(const v2f*)(Br + k);
      acc = __builtin_amdgcn_wmma_f32_16x16x4_f32(
          false, A, false, B, (short)0, acc, false, false);
    }
  }

  // ---- epilogue: deg normalize, history overwrite, store out + his ----
  const int hsize = hsize_p[0];
  const size_t his_off = (size_t)n_node * HID_C;
#pragma unroll
  for (int v = 0; v < 8; ++v) {
    const int m    = v + 8 * hi;       // C/D layout: VGPR v -> M=v (lo lanes) / 8+v (hi lanes)
    const int node = node0 + m;
    if (node < n_node) {
      float o = acc[v] / s_deg[m];
      if (hsize > 0 && s_map[m] != -1) {
        o = hbuf[(size_t)node * HID_C + n0 + ncol];
      }
      const size_t off = (size_t)node * HID_C + n0 + ncol;
      out[off]           = o;
      out[his_off + off] = o;          // his == out
    }
  }
}

extern "C" void kernel_launch(void* const* d_in, const int* in_sizes, int n_in,
                              void* d_out, int out_size, void* d_ws, size_t ws_size,
                              hipStream_t stream) {
  const float* x    = (const float*)d_in[0];
  const float* W    = (const float*)d_in[1];
  const int*   ptr  = (const int*)d_in[2];
  const int*   idx  = (const int*)d_in[3];
  const int*   et   = (const int*)d_in[4];
  const int*   hmap = (const int*)d_in[6];
  const float* hbuf = (const float*)d_in[7];
  const int*   hsz  = (const int*)d_in[9];
  float* out = (float*)d_out;
  float* Wt  = (float*)d_ws;           // 8*64*64 floats = 128 KB scratch

  const int n_node = in_sizes[6];      // history_map length == N_NODE
  const int wtotal = in_sizes[1];      // R*IN_C*HID_C

  wt_transpose_kernel<<<(wtotal + 255) / 256, 256, 0, stream>>>(W, Wt, wtotal);

  const int nblocks = (n_node + TILE_M - 1) / TILE_M;
  rgcn_fused_kernel<<<nblocks, 128, 0, stream>>>(x, Wt, ptr, idx, et, hmap,
                                                 hbuf, hsz, out, n_node);
}